// JpegLayer_84499186581686
// MI455X (gfx1250) — compile-verified
//
#include <hip/hip_runtime.h>
#include <math.h>

typedef float v2f __attribute__((ext_vector_type(2)));
typedef float v8f __attribute__((ext_vector_type(8)));

#define OFFS 0.50196078431372549f   // 128/255
#define QSCL 2.5f                   // QUALITY/100
#define QADD 0.00196078431372549f   // 0.5/255

// 8x8 DCT-II matrix, row-major. 0.5*cos(k*pi/16) family, CD = 1/sqrt(8)
#define CA 0.49039264020161522f
#define CB 0.46193976625564337f
#define CC 0.41573480615127262f
#define CD 0.35355339059327373f
#define CE 0.27778511650980114f
#define CF 0.19134171618254489f
#define CG 0.09754516100806412f
static __device__ __constant__ __align__(16) float D8TAB[64] = {
     CD,  CD,  CD,  CD,  CD,  CD,  CD,  CD,
     CA,  CC,  CE,  CG, -CG, -CE, -CC, -CA,
     CB,  CF, -CF, -CB, -CB, -CF,  CF,  CB,
     CC, -CG, -CA, -CE,  CE,  CA,  CG, -CC,
     CD, -CD, -CD,  CD,  CD, -CD, -CD,  CD,
     CE, -CA,  CG,  CC, -CC, -CG,  CA, -CE,
     CF, -CB,  CB, -CF, -CF,  CB, -CB,  CF,
     CG, -CE,  CC, -CA,  CA, -CC,  CE, -CG
};

__global__ __launch_bounds__(256)
void jpeg_kernel(const float* __restrict__ rgb, const float* __restrict__ quant,
                 float* __restrict__ out, int nb)
{
    __shared__ __align__(16) float Yt[32 * 34];      // 32x32 luma tile, stride 34
    __shared__ __align__(16) float CbT[16 * 18];     // 16x16 chroma, stride 18
    __shared__ __align__(16) float CrT[16 * 18];
    __shared__ __align__(16) float Sc[6 * 16 * 20];  // per-wave transpose scratch, stride 20
    __shared__ float QD[192];                        // reciprocal quant denominators

    const int tid = threadIdx.x;
    const int b   = blockIdx.y;
    const int tr  = blockIdx.x >> 4, tc = blockIdx.x & 15;   // 16x16 tiles of 32x32 px
    const int y0  = tr * 32, x0 = tc * 32;

    // ---------------- stage 1: color convert + 4:2:0 subsample into LDS ----------------
    if (tid < 192)   // 3 channels x 64: reciprocal denominators, one rcp each
        QD[tid] = __builtin_amdgcn_rcpf(quant[tid] * QSCL + QADD);
    {
        const int qy = tid >> 4, qx = tid & 15;   // 2x2 quad per thread
        const int px = x0 + 2 * qx;
        float cbs = 0.f, crs = 0.f;
        #pragma unroll
        for (int dy = 0; dy < 2; ++dy) {
            const int py = y0 + 2 * qy + dy;
            const size_t base = ((size_t)(b * 3) * 512 + py) * 512 + px;
            const v2f rr = __builtin_nontemporal_load((const v2f*)(rgb + base));
            const v2f gg = __builtin_nontemporal_load((const v2f*)(rgb + base + 512 * 512));
            const v2f bb = __builtin_nontemporal_load((const v2f*)(rgb + base + 2 * 512 * 512));
            const float rv[2] = {rr.x, rr.y}, gv[2] = {gg.x, gg.y}, bv[2] = {bb.x, bb.y};
            #pragma unroll
            for (int dx = 0; dx < 2; ++dx) {
                const float R = rv[dx], G = gv[dx], Bv = bv[dx];
                float Y  = fminf(fmaxf(0.299f*R + 0.587f*G + 0.114f*Bv, 0.f), 1.f) - OFFS;
                float Cb = fminf(fmaxf(-0.168735892f*R - 0.331264108f*G + 0.5f*Bv + OFFS, 0.f), 1.f) - OFFS;
                float Cr = fminf(fmaxf(0.5f*R - 0.418687589f*G - 0.081312411f*Bv + OFFS, 0.f), 1.f) - OFFS;
                Yt[(2*qy + dy) * 34 + 2*qx + dx] = Y;
                cbs += Cb; crs += Cr;
            }
        }
        CbT[qy * 18 + qx] = 0.25f * cbs;
        CrT[qy * 18 + qx] = 0.25f * crs;
    }
    __syncthreads();

    // ---------------- stage 2: per-wave 16x16 DCT via f32 WMMA ----------------
    const int w = tid >> 5;          // wave id: 0..3 = Y subtiles, 4 = Cb, 5 = Cr
    if (w < 6) {
        const int l   = tid & 31;
        const int h   = l >> 4;      // lane half
        const int n16 = l & 15;

        // A fragments of block_diag(D8,D8) from constant table
        // (16x4 f32 A-layout: lane L -> M = L%16, K = 2*(L/16)+v)
        v2f a[4];
        #pragma unroll
        for (int c = 0; c < 4; ++c) {
            const int k0 = 4 * c + 2 * h;
            const v2f t = *(const v2f*)&D8TAB[((n16 & 7) << 3) + (k0 & 7)];
            const bool nz = ((n16 >> 3) == (k0 >> 3));   // block-diagonal mask
            a[c].x = nz ? t.x : 0.0f;
            a[c].y = nz ? t.y : 0.0f;
        }

        // source tile in LDS
        const float* src; int ss;
        if (w < 4)       { src = Yt + ((w >> 1) * 16) * 34 + (w & 1) * 16; ss = 34; }
        else if (w == 4) { src = CbT; ss = 18; }
        else             { src = CrT; ss = 18; }

        // B fragments of X^T: bf[c][v][lane] = X[n16][4c+2h+v]
        v2f bf[4];
        #pragma unroll
        for (int c = 0; c < 4; ++c) {
            const int k0 = 4 * c + 2 * h;
            bf[c].x = src[n16 * ss + k0];
            bf[c].y = src[n16 * ss + k0 + 1];
        }

        // T = D16 * X^T  (= (X * D16^T)^T), accumulate K in 4-chunks
        v8f acc = {0.f, 0.f, 0.f, 0.f, 0.f, 0.f, 0.f, 0.f};
        #pragma unroll
        for (int c = 0; c < 4; ++c)
            acc = __builtin_amdgcn_wmma_f32_16x16x4_f32(false, a[c], false, bf[c],
                                                        (short)0, acc, false, false);

        // C/D layout -> row-major LDS scratch (lane l, vgpr r holds T[r+8h][n16])
        float* S = Sc + w * 16 * 20;
        #pragma unroll
        for (int r = 0; r < 8; ++r)
            S[(r + 8 * h) * 20 + n16] = acc[r];

        asm volatile("s_wait_dscnt 0x0" ::: "memory");   // wave-local store->load ordering

        // B fragments of T^T: bb[c][v][lane] = T[n16][4c+2h+v]
        v2f bb[4];
        #pragma unroll
        for (int c = 0; c < 4; ++c) {
            const int k0 = 4 * c + 2 * h;
            bb[c].x = S[n16 * 20 + k0];
            bb[c].y = S[n16 * 20 + k0 + 1];
        }

        // out = D16 * T^T = D16 * X * D16^T  (per-8x8-block 2-D DCT)
        v8f acc2 = {0.f, 0.f, 0.f, 0.f, 0.f, 0.f, 0.f, 0.f};
        #pragma unroll
        for (int c = 0; c < 4; ++c)
            acc2 = __builtin_amdgcn_wmma_f32_16x16x4_f32(false, a[c], false, bb[c],
                                                         (short)0, acc2, false, false);

        // quantize (multiply by precomputed reciprocal), RTE round, NT store
        const int ch = (w < 4) ? 0 : (w - 3);
        const float* qr = QD + ch * 64;
        const size_t yOff  = (size_t)nb * 262144;   // nb*64*64*64
        const size_t cbCnt = (size_t)nb * 65536;    // nb*32*32*64

        size_t base;
        if (w < 4) {
            const int HB = tr * 4 + (w >> 1) * 2 + h;
            const int WB = tc * 4 + (w & 1) * 2 + ((l >> 3) & 1);
            base = (((size_t)b * 64 + HB) * 64 + WB) * 64 + (l & 7);
        } else {
            const int HB = tr * 2 + h;
            const int WB = tc * 2 + ((l >> 3) & 1);
            base = ((w == 4) ? yOff : yOff + cbCnt)
                 + (((size_t)b * 32 + HB) * 32 + WB) * 64 + (l & 7);
        }
        #pragma unroll
        for (int r = 0; r < 8; ++r) {
            const float v = rintf(acc2[r] * qr[r * 8 + (l & 7)]);
            __builtin_nontemporal_store(v, out + base + r * 8);
        }
    }
}

extern "C" void kernel_launch(void* const* d_in, const int* in_sizes, int n_in,
                              void* d_out, int out_size, void* d_ws, size_t ws_size,
                              hipStream_t stream) {
    const float* rgb   = (const float*)d_in[0];
    const float* quant = (const float*)d_in[1];
    float* out = (float*)d_out;
    const int nb = in_sizes[0] / (3 * 512 * 512);   // batch (16)
    dim3 grid(256, nb, 1);                          // 16x16 tiles of 32x32 px per image
    jpeg_kernel<<<grid, 256, 0, stream>>>(rgb, quant, out, nb);
}